// GbldDetrDiscriminativeLoss_70841190580937
// MI455X (gfx1250) — compile-verified
//
#include <hip/hip_runtime.h>
#include <hip/hip_bf16.h>

typedef float v2f __attribute__((ext_vector_type(2)));
typedef float v8f __attribute__((ext_vector_type(8)));

#define B_ 16
#define D_ 16
#define H_ 288
#define W_ 512
#define NPIX (H_*W_)            // 147456 pixels per image
#define DELTA_VAR 0.5f
#define DELTA_DIST 1.5f
#define EPSV 1e-12f

// ws layout (floats): cnt[B][16] | sums[B][16*16] | term[B][16] | means[B][16*16]
#define CNT_OFF   0
#define SUMS_OFF  (B_*16)
#define TERM_OFF  (SUMS_OFF + B_*256)
#define MEANS_OFF (TERM_OFF + B_*16)
#define ZERO_N    MEANS_OFF     // contiguous region that must be zeroed

#define P1_BLOCKS_PER_IMG 32
#define P1_PIX_PER_BLOCK (NPIX / P1_BLOCKS_PER_IMG)   // 4608
#define P1_TILE 256
#define P1_TILES (P1_PIX_PER_BLOCK / P1_TILE)         // 18

#define P3_BLOCKS_PER_IMG 64
#define P3_PIX (NPIX / P3_BLOCKS_PER_IMG)             // 2304

// ---- CDNA5 async global->LDS helpers (ASYNCcnt-tracked DMA, inline asm) ----
__device__ __forceinline__ unsigned ldsOff(const void* p) {
    // addrspace(3) pointers are the raw 32-bit LDS offset
    return (unsigned)(unsigned long long)(const __attribute__((address_space(3))) char*)p;
}
__device__ __forceinline__ void asyncLds128(unsigned ldsdst, unsigned gofs, const void* sbase) {
    asm volatile("global_load_async_to_lds_b128 %0, %1, %2"
                 :: "v"(ldsdst), "v"(gofs), "s"(sbase) : "memory");
}
__device__ __forceinline__ void asyncLds32(unsigned ldsdst, unsigned gofs, const void* sbase) {
    asm volatile("global_load_async_to_lds_b32 %0, %1, %2"
                 :: "v"(ldsdst), "v"(gofs), "s"(sbase) : "memory");
}
__device__ __forceinline__ void waitAsync0() {
    asm volatile("s_wait_asynccnt 0x0" ::: "memory");
}

__global__ __launch_bounds__(256)
void zero_k(float* __restrict__ ws, float* __restrict__ out, int n) {
    int i = blockIdx.x * blockDim.x + threadIdx.x;
    if (i < n) ws[i] = 0.0f;
    if (i == 0) out[0] = 0.0f;
}

// -------- Pass 1: per-label counts + embedding sums via onehot-WMMA GEMM --------
// sums[l][d] = sum_pix onehot(seg==l+1) * e[pix][d], accumulated in each wave's
// 16x16 f32 WMMA accumulator (labels x dims), K=4 pixels per v_wmma_f32_16x16x4_f32.
// Tiles are double-buffered: async DMA of tile T+1 overlaps WMMA compute of tile T.
__global__ __launch_bounds__(256)
void pass1_k(const float* __restrict__ emb, const int* __restrict__ seg,
             float* __restrict__ ws) {
    __shared__ __align__(16) float se[2][D_ * 260];  // pitch 260 -> conflict-free col reads
    __shared__ __align__(16) int   ss[2][P1_TILE];
    __shared__ float sums_s[256];
    __shared__ float cnt_s[16];

    const int t     = threadIdx.x;
    const int b     = blockIdx.x / P1_BLOCKS_PER_IMG;
    const int chunk = blockIdx.x % P1_BLOCKS_PER_IMG;
    const int pix0  = chunk * P1_PIX_PER_BLOCK;
    const int wave  = t >> 5;
    const int lane  = t & 31;
    const int j     = lane & 15;   // label index (A) / dim index (B)
    const int half  = lane >> 4;

    sums_s[t] = 0.0f;
    if (t < 16) cnt_s[t] = 0.0f;

    // uniform global bases (SGPR), per-thread 32-bit byte offsets (VGPR)
    const int ld = t >> 4;   // dim row this thread DMAs
    const int lg = t & 15;   // 16-float column group
    const float* egbase = emb + (size_t)b * D_ * NPIX;
    const int*   sgbase = seg + (size_t)b * NPIX;
    const unsigned erow = (unsigned)((ld * NPIX + pix0 + lg * 16) * 4);
    const unsigned srow = (unsigned)((pix0 + t) * 4);
    unsigned edst[2], sdst[2];
    edst[0] = ldsOff(&se[0][ld * 260 + lg * 16]);
    edst[1] = ldsOff(&se[1][ld * 260 + lg * 16]);
    sdst[0] = ldsOff(&ss[0][t]);
    sdst[1] = ldsOff(&ss[1][t]);

    // prologue: DMA tile 0 into buffer 0
    #pragma unroll
    for (int k = 0; k < 4; ++k)
        asyncLds128(edst[0] + k * 16, erow + k * 16, egbase);
    asyncLds32(sdst[0], srow, sgbase);

    v8f  c    = {};
    float cacc = 0.0f;

    for (int tile = 0; tile < P1_TILES; ++tile) {
        waitAsync0();        // this wave's DMAs for buf[tile&1] have landed
        __syncthreads();     // everyone's landed; prior readers of other buf done

        if (tile + 1 < P1_TILES) {   // kick DMA for next tile while we compute
            const int nb = (tile + 1) & 1;
            const unsigned go = (unsigned)((tile + 1) * P1_TILE * 4);
            #pragma unroll
            for (int k = 0; k < 4; ++k)
                asyncLds128(edst[nb] + k * 16, erow + go + k * 16, egbase);
            asyncLds32(sdst[nb], srow + go, sgbase);
        }

        const float* sep = se[tile & 1];
        const int*   ssp = ss[tile & 1];
        const int qb = wave * 32;    // this wave's 32-pixel slice of the tile
        #pragma unroll
        for (int stp = 0; stp < 8; ++stp) {
            const int q = qb + stp * 4;
            // B: lane j holds e[dim=j][pix], K0/K1 lanes0-15, K2/K3 lanes16-31
            v2f bm = *(const v2f*)(sep + j * 260 + q + 2 * half);
            int2 sv = *(const int2*)(ssp + q + 2 * half);
            float ax = (sv.x == j + 1) ? 1.0f : 0.0f;
            float ay = (sv.y == j + 1) ? 1.0f : 0.0f;
            cacc += ax + ay;                       // free per-label pixel counts
            v2f a = { ax, ay };                    // onehot A, same K split
            c = __builtin_amdgcn_wmma_f32_16x16x4_f32(
                    false, a, false, bm, (short)0, c, false, false);
        }
    }

    // block reduction: C tile -> LDS -> one global atomic per thread
    #pragma unroll
    for (int r = 0; r < 8; ++r)
        atomicAdd(&sums_s[(r + 8 * half) * 16 + j], c[r]);
    atomicAdd(&cnt_s[j], cacc);
    __syncthreads();
    atomicAdd(&ws[SUMS_OFF + b * 256 + t], sums_s[t]);
    if (t < 16) atomicAdd(&ws[CNT_OFF + b * 16 + t], cnt_s[t]);
}

// -------- Pass 2: means = sums / max(cnt,1) --------
__global__ __launch_bounds__(256)
void means_k(float* __restrict__ ws) {
    const int b = blockIdx.x, t = threadIdx.x;
    const float cnt = ws[CNT_OFF + b * 16 + (t >> 4)];
    ws[MEANS_OFF + b * 256 + t] = ws[SUMS_OFF + b * 256 + t] / fmaxf(cnt, 1.0f);
}

// -------- Pass 3: variance (pull) term, per-label term sums --------
__global__ __launch_bounds__(256)
void pass3_k(const float* __restrict__ emb, const int* __restrict__ seg,
             float* __restrict__ ws) {
    __shared__ float m_s[16 * 17];   // pitch 17 -> distinct-s lanes hit distinct banks
    __shared__ float t_s[16];
    const int t     = threadIdx.x;
    const int b     = blockIdx.x / P3_BLOCKS_PER_IMG;
    const int chunk = blockIdx.x % P3_BLOCKS_PER_IMG;
    const int p0    = chunk * P3_PIX;

    if (t < 16) t_s[t] = 0.0f;
    m_s[(t >> 4) * 17 + (t & 15)] = ws[MEANS_OFF + b * 256 + t];
    __syncthreads();

    const float* eb = emb + (size_t)b * D_ * NPIX;
    const int*   sb = seg + (size_t)b * NPIX;
    for (int k = 0; k < P3_PIX / 256; ++k) {
        const int n = p0 + k * 256 + t;   // coalesced across threads
        const int s = sb[n];
        if (s > 0) {                       // label-0 terms are masked out anyway
            const float* mrow = m_s + (s - 1) * 17;
            float acc = 0.0f;
            #pragma unroll
            for (int d = 0; d < 16; ++d) {
                const float diff = eb[(size_t)d * NPIX + n] - mrow[d];
                acc = fmaf(diff, diff, acc);
            }
            const float nrm = sqrtf(fmaxf(acc, EPSV));
            const float r   = fmaxf(nrm - DELTA_VAR, 0.0f);
            atomicAdd(&t_s[s - 1], r * r);
        }
    }
    __syncthreads();
    if (t < 16) atomicAdd(&ws[TERM_OFF + b * 16 + t], t_s[t]);
}

// -------- Finalize: centroid Gram via WMMA, push + var terms, scalar loss --------
__global__ __launch_bounds__(32)
void fin_k(const float* __restrict__ ws, float* __restrict__ out) {
    __shared__ float gmat[256];
    __shared__ float pres_s[16];
    const int lane = threadIdx.x;
    const int b    = blockIdx.x;
    const int j    = lane & 15;
    const int half = lane >> 4;

    const float cnt = ws[CNT_OFF + b * 16 + j];
    const int present = cnt > 0.0f;
    const unsigned long long bm = __ballot(present);
    const float nl = (float)__popcll(bm & 0xFFFFull);

    // G = M * M^T : per-lane A (row j) and B (col j) values are identical
    v8f g = {};
    const float* mb = ws + MEANS_OFF + b * 256;
    #pragma unroll
    for (int kk = 0; kk < 4; ++kk) {
        v2f a = *(const v2f*)(mb + j * 16 + kk * 4 + 2 * half);
        g = __builtin_amdgcn_wmma_f32_16x16x4_f32(
                false, a, false, a, (short)0, g, false, false);
    }
    #pragma unroll
    for (int r = 0; r < 8; ++r) gmat[(r + 8 * half) * 16 + j] = g[r];
    if (half == 0) pres_s[j] = present ? 1.0f : 0.0f;
    __syncthreads();

    float var_p = 0.0f;
    if (half == 0 && present) var_p = ws[TERM_OFF + b * 16 + j] / cnt;

    float push = 0.0f;
    const float gdj = gmat[j * 16 + j];
    const float pj  = pres_s[j];
    #pragma unroll
    for (int r = 0; r < 8; ++r) {
        const int l1 = r + 8 * half;
        if (l1 != j) {
            const float msk = pres_s[l1] * pj;
            const float ssq = fmaxf(gmat[l1 * 16 + l1] + gdj - 2.0f * g[r], 0.0f);
            const float rr  = fmaxf(DELTA_DIST - sqrtf(ssq), 0.0f);
            push += msk * rr * rr;
        }
    }
    #pragma unroll
    for (int off = 16; off > 0; off >>= 1) {
        var_p += __shfl_xor(var_p, off, 32);
        push  += __shfl_xor(push,  off, 32);
    }
    if (lane == 0) {
        const float var_l  = (nl > 0.0f) ? var_p / nl : 0.0f;
        const float denom  = fmaxf(nl * (nl - 1.0f), 1.0f);
        const float dist_l = (nl > 1.0f) ? push / denom * 0.5f : 0.0f;
        atomicAdd(out, (var_l + dist_l) * (1.0f / (float)B_));
    }
}

extern "C" void kernel_launch(void* const* d_in, const int* in_sizes, int n_in,
                              void* d_out, int out_size, void* d_ws, size_t ws_size,
                              hipStream_t stream) {
    const float* emb = (const float*)d_in[0];
    const int*   seg = (const int*)d_in[1];
    float* out = (float*)d_out;
    float* ws  = (float*)d_ws;

    zero_k <<<(ZERO_N + 255) / 256, 256, 0, stream>>>(ws, out, ZERO_N);
    pass1_k<<<B_ * P1_BLOCKS_PER_IMG, 256, 0, stream>>>(emb, seg, ws);
    means_k<<<B_, 256, 0, stream>>>(ws);
    pass3_k<<<B_ * P3_BLOCKS_PER_IMG, 256, 0, stream>>>(emb, seg, ws);
    fin_k  <<<B_, 32, 0, stream>>>(ws, out);
}